// DepthWiseConv1d_25211458027710
// MI455X (gfx1250) — compile-verified
//
#include <hip/hip_runtime.h>

typedef float v4f __attribute__((ext_vector_type(4)));

#define TILE    2048
#define THREADS 256

// LDS layout: [3] = left halo, [4 .. 4+TILE-1] = x tile, [4+TILE] = right halo.
// Tile starts at byte offset 16 -> every async B128 destination is 16B aligned.
__global__ __launch_bounds__(THREADS) void dwconv1d_k3_kernel(
    const float* __restrict__ x, const float* __restrict__ w,
    const float* __restrict__ b, float* __restrict__ out,
    int C, int L)
{
    __shared__ __align__(16) float smem[TILE + 8];

    const int      t    = threadIdx.x;
    const unsigned blk  = blockIdx.x;
    const unsigned row  = blk >> 1;        // row = bIdx*C + c
    const int      half = blk & 1;
    const int      l0   = half * TILE;
    const int      c    = (int)(row & (unsigned)(C - 1));   // C = 512, pow2
    const size_t   base = (size_t)row * (size_t)L;
    const float* __restrict__ xrow = x + base;

    // Each thread owns two lane-contiguous float4 groups: e0 in [0,1024),
    // e1 in [1024,2048). Within a wave32 that makes each async/ds/global
    // access a contiguous 512B burst.
    const int e0 = t * 4;
    const int e1 = (TILE / 2) + t * 4;

#if defined(__gfx1250__)
    {
        // Low 32 bits of a generic LDS pointer == LDS byte offset (aperture
        // layout, ISA 10.2), which is what VDST of the async op expects.
        unsigned lds0 = (unsigned)(uintptr_t)(&smem[4 + e0]);
        unsigned lds1 = (unsigned)(uintptr_t)(&smem[4 + e1]);
        unsigned long long g0 = (unsigned long long)(uintptr_t)(xrow + l0 + e0);
        unsigned long long g1 = (unsigned long long)(uintptr_t)(xrow + l0 + e1);
        // HBM -> LDS DMA, tracked on ASYNCcnt (no VGPR round trip).
        asm volatile("global_load_async_to_lds_b128 %0, %1, off"
                     :: "v"(lds0), "v"(g0) : "memory");
        asm volatile("global_load_async_to_lds_b128 %0, %1, off"
                     :: "v"(lds1), "v"(g1) : "memory");
    }
#else
    *(v4f*)&smem[4 + e0] = *(const v4f*)(xrow + l0 + e0);
    *(v4f*)&smem[4 + e1] = *(const v4f*)(xrow + l0 + e1);
#endif

    // Halo (zero padding at row edges) via plain loads + ds stores.
    if (t == 0) smem[3]        = (l0 > 0)       ? xrow[l0 - 1]    : 0.0f;
    if (t == 1) smem[4 + TILE] = (l0 + TILE < L) ? xrow[l0 + TILE] : 0.0f;

#if defined(__gfx1250__)
    asm volatile("s_wait_asynccnt 0" ::: "memory");   // this wave's DMA done
#endif
    __syncthreads();                                  // all waves' tiles + halos

    // Block-uniform per-channel taps -> scalar loads.
    const float w0   = w[c * 3 + 0];
    const float w1   = w[c * 3 + 1];
    const float w2   = w[c * 3 + 2];
    const float bias = b[c];

    float* __restrict__ orow = out + base + l0;

    #pragma unroll
    for (int g = 0; g < 2; ++g) {
        const int   e   = (g == 0) ? e0 : e1;
        const int   idx = e + 4;
        const float xl  = smem[idx - 1];
        const v4f   m   = *(const v4f*)&smem[idx];    // ds_load_b128, conflict-free
        const float xr  = smem[idx + 4];

        v4f o;
        o.x = fmaf(w2, m.y, fmaf(w1, m.x, fmaf(w0, xl,  bias)));
        o.y = fmaf(w2, m.z, fmaf(w1, m.y, fmaf(w0, m.x, bias)));
        o.z = fmaf(w2, m.w, fmaf(w1, m.z, fmaf(w0, m.y, bias)));
        o.w = fmaf(w2, xr,  fmaf(w1, m.w, fmaf(w0, m.z, bias)));

        // Write-once output: non-temporal so 134MB of stores don't evict
        // x halo lines from L2.
        __builtin_nontemporal_store(o, (v4f*)(orow + e));
    }
}

extern "C" void kernel_launch(void* const* d_in, const int* in_sizes, int n_in,
                              void* d_out, int out_size, void* d_ws, size_t ws_size,
                              hipStream_t stream) {
    const float* x = (const float*)d_in[0];   // [16, 512, 4096]
    const float* w = (const float*)d_in[1];   // [512, 1, 3]
    const float* b = (const float*)d_in[2];   // [512]
    float* out = (float*)d_out;               // [16, 512, 4096]

    const int B = 16, C = 512, L = 4096;
    const int blocks = B * C * (L / TILE);    // 16384 blocks, 8 wave32 each

    dwconv1d_k3_kernel<<<blocks, THREADS, 0, stream>>>(x, w, b, out, C, L);
}